// MMDLoss_10376640987957
// MI455X (gfx1250) — compile-verified
//
#include <hip/hip_runtime.h>

typedef float v2f __attribute__((ext_vector_type(2)));
typedef float v8f __attribute__((ext_vector_type(8)));
typedef int   v4i __attribute__((vector_size(16)));
typedef __attribute__((address_space(1))) v4i* gv4i_p;   // global int4*
typedef __attribute__((address_space(3))) v4i* lv4i_p;   // LDS int4*

#define N_S   4096
#define NTOT  8192
#define DDIM  256
#define KC    32            // K-chunk staged in LDS
#define ASTR  36            // padded LDS row stride (floats): 16B-aligned rows, bank-conflict-free
#define NCHUNK (DDIM / KC)  // 8

#if defined(__has_builtin)
#  if __has_builtin(__builtin_amdgcn_global_load_async_to_lds_b128)
#    define HAVE_ASYNC_LDS 1
#  endif
#  if __has_builtin(__builtin_amdgcn_s_wait_asynccnt)
#    define HAVE_WAIT_ASYNC 1
#  endif
#endif

__device__ __forceinline__ void async_copy16(const float* g, float* l) {
#if defined(HAVE_ASYNC_LDS)
    __builtin_amdgcn_global_load_async_to_lds_b128((gv4i_p)g, (lv4i_p)l, 0, 0);
#else
    unsigned lofs = (unsigned)(uintptr_t)(__attribute__((address_space(3))) void*)l;
    asm volatile("global_load_async_to_lds_b128 %0, %1, off"
                 :: "v"(lofs), "v"(g) : "memory");
#endif
}

__device__ __forceinline__ void wait_async0() {
#if defined(HAVE_WAIT_ASYNC)
    __builtin_amdgcn_s_wait_asynccnt(0);
#else
    asm volatile("s_wait_asynccnt 0x0" ::: "memory");
#endif
}

__device__ __forceinline__ const float* row_ptr(const float* __restrict__ src,
                                                const float* __restrict__ tgt,
                                                int row) {
    return (row < N_S) ? (src + (size_t)row * DDIM)
                       : (tgt + (size_t)(row - N_S) * DDIM);
}

// ---- Kernel A: per-row sum of squares (one wave per row) --------------------
__global__ void __launch_bounds__(256)
k_rowsq(const float* __restrict__ src, const float* __restrict__ tgt,
        float* __restrict__ sq) {
    const int wave = threadIdx.x >> 5;
    const int lane = threadIdx.x & 31;
    const int row  = blockIdx.x * 8 + wave;
    const float* p = row_ptr(src, tgt, row);
    float s = 0.f;
#pragma unroll
    for (int j = 0; j < 8; ++j) { float v = p[lane + 32 * j]; s += v * v; }
#pragma unroll
    for (int off = 16; off >= 1; off >>= 1) s += __shfl_xor(s, off, 32);
    if (lane == 0) sq[row] = s;
}

// ---- Kernel B: partial column sums (block b covers 256 rows) ----------------
__global__ void __launch_bounds__(256)
k_colsum(const float* __restrict__ src, const float* __restrict__ tgt,
         float* __restrict__ part) {
    const int c  = threadIdx.x;
    const int r0 = blockIdx.x * 256;
    float s = 0.f;
    for (int r = r0; r < r0 + 256; ++r) s += row_ptr(src, tgt, r)[c];
    part[blockIdx.x * 256 + c] = s;
}

// ---- Kernel C: bandwidth scalar  inv16B = 1/(16*B);  zero d_out -------------
// sum(L2) = 2*n*sum(sq) - 2*||colsum||^2 ;  B = sum(L2)/(n^2-n)/4
__global__ void __launch_bounds__(256)
k_scalars(const float* __restrict__ sq, const float* __restrict__ part,
          float* __restrict__ scal, float* __restrict__ out) {
    __shared__ float red[256];
    const int t = threadIdx.x;

    float s = 0.f;
    for (int j = 0; j < 32; ++j) s += sq[t + 256 * j];
    red[t] = s;
    __syncthreads();
    for (int off = 128; off >= 1; off >>= 1) {
        if (t < off) red[t] += red[t + off];
        __syncthreads();
    }
    float sumSq = red[0];
    __syncthreads();

    float col = 0.f;
    for (int b = 0; b < 32; ++b) col += part[b * 256 + t];
    red[t] = col * col;
    __syncthreads();
    for (int off = 128; off >= 1; off >>= 1) {
        if (t < off) red[t] += red[t + off];
        __syncthreads();
    }

    if (t == 0) {
        const float n = (float)NTOT;
        float sumL2   = 2.0f * n * sumSq - 2.0f * red[0];
        float bw      = sumL2 / (n * n - n) * 0.25f;  // / KERNEL_MUL^(5//2)
        scal[0]       = 1.0f / (16.0f * bw);          // 1/(16*B)
        out[0]        = 0.0f;                          // init accumulator
    }
}

// ---- Kernel D: async-LDS double-buffered WMMA f32 Gram + exp epilogue -------
// Block: 8 waves -> 128(M) x 64(N) macro-tile. Wave: 16x64 strip (1 A-frag, 4 B-frags).
__global__ void __launch_bounds__(256)
k_mmd(const float* __restrict__ src, const float* __restrict__ tgt,
      const float* __restrict__ sq, const float* __restrict__ scal,
      float* __restrict__ out) {
    __shared__ float As[2][128 * ASTR];   // 36 KB
    __shared__ float Bs[2][64 * ASTR];    // 18 KB
    __shared__ float red[8];

    const int tid  = threadIdx.x;
    const int lane = tid & 31;
    const int wave = tid >> 5;
    const int mblk = blockIdx.x * 128;
    const int nblk = blockIdx.y * 64;
    const float inv16B = scal[0];

    // Fragment addressing per ISA 16x4 f32 layout:
    // lanes 0-15: M=lane, K=k+{0,1}; lanes 16-31: M=lane-16, K=k+{2,3}
    const int fr   = lane & 15;
    const int koff = (lane >> 4) * 2;

    // Per-thread staging slots: A panel 128x32 = 1024 xfers of 16B -> 4/thread,
    //                           B panel  64x32 =  512 xfers of 16B -> 2/thread.
    const int ra[4] = { (tid + 0) >> 3, (tid + 256) >> 3, (tid + 512) >> 3, (tid + 768) >> 3 };
    const int rb[2] = { (tid + 0) >> 3, (tid + 256) >> 3 };
    const int seg   = (tid & 7) * 4;     // float offset within the 32-float chunk row

    const float* gA[4];
    const float* gB[2];
#pragma unroll
    for (int i = 0; i < 4; ++i) gA[i] = row_ptr(src, tgt, mblk + ra[i]) + seg;
#pragma unroll
    for (int i = 0; i < 2; ++i) gB[i] = row_ptr(src, tgt, nblk + rb[i]) + seg;

    auto async_chunk = [&](int c, int buf) {
        const int k0 = c * KC;
#pragma unroll
        for (int i = 0; i < 4; ++i)
            async_copy16(gA[i] + k0, &As[buf][ra[i] * ASTR + seg]);
#pragma unroll
        for (int i = 0; i < 2; ++i)
            async_copy16(gB[i] + k0, &Bs[buf][rb[i] * ASTR + seg]);
    };

    v8f acc[4];
#pragma unroll
    for (int t = 0; t < 4; ++t) acc[t] = (v8f){0, 0, 0, 0, 0, 0, 0, 0};

    async_chunk(0, 0);

    for (int c = 0; c < NCHUNK; ++c) {
        wait_async0();                             // this wave's chunk-c xfers landed
        __syncthreads();                           // all waves': buf ready, buf^1 free
        if (c + 1 < NCHUNK) async_chunk(c + 1, (c + 1) & 1);  // overlaps WMMAs below

        const int buf = c & 1;
        const float* aw = &As[buf][(wave * 16 + fr) * ASTR];
#pragma unroll
        for (int k = 0; k < KC; k += 4) {
            v2f a = *(const v2f*)(aw + k + koff);
#pragma unroll
            for (int t = 0; t < 4; ++t) {
                v2f b = *(const v2f*)(&Bs[buf][(16 * t + fr) * ASTR + k + koff]);
                acc[t] = __builtin_amdgcn_wmma_f32_16x16x4_f32(
                    false, a, false, b, (short)0, acc[t], false, false);
            }
        }
    }

    // C/D layout: lanes 0-15 -> M=v, lanes 16-31 -> M=v+8; N = lane&15
    const int gm_base = mblk + wave * 16 + (lane >> 4) * 8;
    float accs = 0.f;
#pragma unroll
    for (int t = 0; t < 4; ++t) {
        const int   gn  = nblk + 16 * t + (lane & 15);
        const float sqn = sq[gn];
        const bool  bn  = gn < N_S;
#pragma unroll
        for (int v = 0; v < 8; ++v) {
            const int gm = gm_base + v;
            float l2 = sq[gm] + sqn - 2.0f * acc[t][v];
            float x  = -l2 * inv16B;
            float y  = __expf(x);               // y = exp(-L2/(16B))
            float y2 = y * y, y4 = y2 * y2, y8 = y4 * y4, y16 = y8 * y8;
            float s5 = y + y2 + y4 + y8 + y16;  // 5-kernel sum
            const bool bm = gm < N_S;
            accs += (bm == bn) ? s5 : -s5;
        }
    }

#pragma unroll
    for (int off = 16; off >= 1; off >>= 1) accs += __shfl_xor(accs, off, 32);
    if (lane == 0) red[wave] = accs;
    __syncthreads();
    if (tid == 0) {
        float s = 0.f;
#pragma unroll
        for (int w = 0; w < 8; ++w) s += red[w];
        atomicAdd(out, s * (1.0f / ((float)N_S * (float)N_S)));
    }
}

extern "C" void kernel_launch(void* const* d_in, const int* in_sizes, int n_in,
                              void* d_out, int out_size, void* d_ws, size_t ws_size,
                              hipStream_t stream) {
    (void)in_sizes; (void)n_in; (void)out_size; (void)ws_size;
    const float* src = (const float*)d_in[0];
    const float* tgt = (const float*)d_in[1];
    float* out  = (float*)d_out;
    float* SQ   = (float*)d_ws;          // 8192 floats
    float* PART = SQ + NTOT;             // 32*256 floats
    float* SCAL = PART + 32 * 256;       // 1 float

    k_rowsq  <<<NTOT / 8, 256, 0, stream>>>(src, tgt, SQ);
    k_colsum <<<32,       256, 0, stream>>>(src, tgt, PART);
    k_scalars<<<1,        256, 0, stream>>>(SQ, PART, SCAL, out);

    dim3 grid(NTOT / 128, NTOT / 64);    // (64, 128) blocks of 8 waves
    k_mmd    <<<grid,     256, 0, stream>>>(src, tgt, SQ, SCAL, out);
}